// PathEncoder_12111807775260
// MI455X (gfx1250) — compile-verified
//
#include <hip/hip_runtime.h>
#include <hip/hip_bf16.h>

// Problem constants (from the reference)
#define B_  8
#define N_  128
#define L_  5
#define D_  2
#define NE_ 513        // NUM_EDGES + 1 rows in edge_emb
#define F_  64
#define H_  8

#define NTILES ((NE_ + 15) / 16)        // 33 row-tiles of 16 edges
#define W2_ELEMS (L_ * NE_ * H_)        // 20520 floats = 82080 bytes
#define W2_VEC4  (W2_ELEMS / 4)         // 5130 float4s (82080 % 16 == 0)

typedef float v2f __attribute__((ext_vector_type(2)));
typedef float v8f __attribute__((ext_vector_type(8)));

// ---------------------------------------------------------------------------
// Kernel 1: W2[l][e][h] = sum_f edge_emb[e][f] * attn_w[l][f][h]
// One wave32 per (l, 16-edge tile). V_WMMA_F32_16X16X4_F32 chained over K=64.
//   A (16x4 f32):  lanes 0-15 -> M=lane, VGPR0/1 = K=0/1 ; lanes 16-31 -> K=2/3
//   B (4x16 f32):  lanes 0-15 -> N=lane, VGPR0/1 = K=0/1 ; lanes 16-31 -> K=2/3
//   C/D (16x16):   VGPR r: lanes 0-15 -> M=r, N=lane ; lanes 16-31 -> M=8+r
// Out-of-range rows/cols are CLAMPED (not predicated): they only produce
// garbage C entries that are never stored, so all loads are unconditional and
// EXEC stays all-ones with zero exec-mask traffic.
// ---------------------------------------------------------------------------
__global__ __launch_bounds__(32)
void path_enc_wmma_precompute(const float* __restrict__ edge_emb,
                              const float* __restrict__ attn_w,
                              float* __restrict__ W2)
{
    const int l    = blockIdx.x / NTILES;
    const int tile = blockIdx.x % NTILES;
    const int lane = threadIdx.x;            // 0..31, full wave -> EXEC all 1s
    const int m     = lane & 15;             // A row within tile / B,C column
    const int khalf = (lane >> 4) * 2;       // 0 for lanes 0-15, 2 for 16-31

    const int row  = tile * 16 + m;
    const int rowc = min(row, NE_ - 1);      // clamp: garbage rows never stored
    const int colc = min(m, H_ - 1);         // clamp: garbage cols never stored

    const float* __restrict__ ap = edge_emb + rowc * F_;           // 256B-aligned
    const float* __restrict__ bp = attn_w + l * (F_ * H_) + colc;

    v8f c = {};
#pragma unroll
    for (int k0 = 0; k0 < F_; k0 += 4) {
        const v2f a = *(const v2f*)(ap + k0 + khalf);   // one b64 load
        v2f b;
        b.x = bp[(k0 + khalf + 0) * H_];
        b.y = bp[(k0 + khalf + 1) * H_];
        c = __builtin_amdgcn_wmma_f32_16x16x4_f32(
                /*neg_a=*/false, a, /*neg_b=*/false, b,
                /*c_mod=*/(short)0, c, /*reuse_a=*/false, /*reuse_b=*/false);
    }

    if (m < H_) {
        const int mbase = tile * 16 + ((lane >> 4) ? 8 : 0);
#pragma unroll
        for (int r = 0; r < 8; ++r) {
            const int rr = mbase + r;
            if (rr < NE_) W2[(l * NE_ + rr) * H_ + m] = c[r];
        }
    }
}

// ---------------------------------------------------------------------------
// Kernel 2: bandwidth kernel.
//   out[g][h] = (1 / (2*dist[g])) * sum_{l,d} W2[l][path[g][l][d]][h]
// W2 (82 KB) staged into LDS via float4 copies; each thread handles one
// (b,x,y) cell: 10 index loads (b64-pairable), 10 x 2 ds_load_b128 row
// gathers, 2 global_store_b128. Pure HBM-bandwidth shape (~10 MB traffic).
// ---------------------------------------------------------------------------
__global__ __launch_bounds__(256)
void path_enc_gather(const int* __restrict__ dist,
                     const int* __restrict__ path,
                     const float4* __restrict__ W2v,
                     float* __restrict__ out)
{
    extern __shared__ float lds[];   // W2_ELEMS floats
    float4* __restrict__ ldsv = (float4*)lds;
    for (int i = threadIdx.x; i < W2_VEC4; i += blockDim.x)
        ldsv[i] = W2v[i];
    __syncthreads();

    const int gid = blockIdx.x * blockDim.x + threadIdx.x;   // < B*N*N

    const int* __restrict__ p = path + gid * (L_ * D_);

    float acc[H_];
#pragma unroll
    for (int h = 0; h < H_; ++h) acc[h] = 0.0f;

#pragma unroll
    for (int l = 0; l < L_; ++l) {
#pragma unroll
        for (int d = 0; d < D_; ++d) {
            const int e = p[l * D_ + d];
            const float4* __restrict__ row =
                (const float4*)&lds[(l * NE_ + e) * H_];   // 32B-aligned
            const float4 r0 = row[0];
            const float4 r1 = row[1];
            acc[0] += r0.x; acc[1] += r0.y; acc[2] += r0.z; acc[3] += r0.w;
            acc[4] += r1.x; acc[5] += r1.y; acc[6] += r1.z; acc[7] += r1.w;
        }
    }

    const float inv = 1.0f / ((float)D_ * (float)dist[gid]);

    float4 o0, o1;
    o0.x = acc[0] * inv; o0.y = acc[1] * inv; o0.z = acc[2] * inv; o0.w = acc[3] * inv;
    o1.x = acc[4] * inv; o1.y = acc[5] * inv; o1.z = acc[6] * inv; o1.w = acc[7] * inv;

    float4* __restrict__ op = (float4*)(out + (size_t)gid * H_);
    op[0] = o0;
    op[1] = o1;
}

// ---------------------------------------------------------------------------
extern "C" void kernel_launch(void* const* d_in, const int* in_sizes, int n_in,
                              void* d_out, int out_size, void* d_ws, size_t ws_size,
                              hipStream_t stream)
{
    const int*   dist     = (const int*)d_in[0];     // [B,N,N]
    const int*   path     = (const int*)d_in[1];     // [B,N,N,L,D]
    const float* edge_emb = (const float*)d_in[2];   // [NE, F]
    const float* attn_w   = (const float*)d_in[3];   // [L, F, H]
    float*       out      = (float*)d_out;           // [B,N,N,H]
    float*       W2       = (float*)d_ws;            // [L, NE, H] scratch

    // Stage 1: tiny WMMA GEMM -> W2 table (one wave per (l, tile))
    path_enc_wmma_precompute<<<L_ * NTILES, 32, 0, stream>>>(edge_emb, attn_w, W2);

    // Stage 2: bandwidth gather/normalize (same stream -> ordered after stage 1)
    const int cells  = B_ * N_ * N_;                 // 131072
    const int tpb    = 256;
    const int blocks = cells / tpb;                  // 512
    const size_t smem = (size_t)W2_ELEMS * sizeof(float);  // 82080 B of LDS
    path_enc_gather<<<blocks, tpb, smem, stream>>>(dist, path, (const float4*)W2, out);
}